// DenseNGCNLayer_13357348290975
// MI455X (gfx1250) — compile-verified
//
#include <hip/hip_runtime.h>

typedef float v2f __attribute__((ext_vector_type(2)));
typedef float v4f __attribute__((ext_vector_type(4)));
typedef float v8f __attribute__((ext_vector_type(8)));

#define IN_C 128
#define OUT_C 64

// ---------------------------------------------------------------------------
// Kernel 1: dense projection  base = F[N,128] @ W[128,64]  via fp32 WMMA.
// One wave (32 lanes) computes a 16x64 output tile: four 16x16 accumulators,
// K swept 0..128 in steps of 4 with V_WMMA_F32_16X16X4_F32.
//
// A-fragment layout (16x4 f32): lanes 0-15 hold M=lane, VGPR0=K0,VGPR1=K1;
//                               lanes 16-31 hold M=lane-16, VGPR0=K2,VGPR1=K3.
// B-fragment layout (4x16 f32): lanes 0-15 N=lane K0/K1; lanes 16-31 N=lane-16 K2/K3.
// C/D layout (16x16 f32, 8 VGPRs): lanes 0-15 -> M=v, N=lane;
//                                  lanes 16-31 -> M=v+8, N=lane-16.
// ---------------------------------------------------------------------------
__global__ __launch_bounds__(128) void ngcn_gemm_wmma(
    const float* __restrict__ F, const float* __restrict__ W,
    float* __restrict__ out, int N, int nRowBlocks)
{
    const int wave  = threadIdx.x >> 5;
    const int lane  = threadIdx.x & 31;
    const int rb    = blockIdx.x * 4 + wave;
    if (rb >= nRowBlocks) return;           // whole-wave uniform exit: EXEC stays all-1 for WMMA

    const int m     = lane & 15;            // row (A) / col (B,C) within tile
    const int khalf = lane >> 4;            // 0 -> K{0,1}, 1 -> K{2,3}

    int arow = rb * 16 + m;
    if (arow >= N) arow = N - 1;            // clamp tail loads (store is predicated)

    const float* __restrict__ fp = F + (size_t)arow * IN_C + 2 * khalf;
    const float* __restrict__ wp = W + (size_t)(2 * khalf) * OUT_C + m;

    v8f acc0 = {}, acc1 = {}, acc2 = {}, acc3 = {};

    #pragma unroll 4
    for (int kk = 0; kk < IN_C; kk += 4) {
        // A fragment: two consecutive K values for this lane (8B aligned load)
        v2f a = *(const v2f*)(fp + kk);

        const float* wk = wp + (size_t)kk * OUT_C;
        v2f b0, b1, b2, b3;
        b0.x = wk[ 0];        b0.y = wk[ 0 + OUT_C];
        b1.x = wk[16];        b1.y = wk[16 + OUT_C];
        b2.x = wk[32];        b2.y = wk[32 + OUT_C];
        b3.x = wk[48];        b3.y = wk[48 + OUT_C];

        acc0 = __builtin_amdgcn_wmma_f32_16x16x4_f32(false, a, false, b0, (short)0, acc0, false, false);
        acc1 = __builtin_amdgcn_wmma_f32_16x16x4_f32(false, a, false, b1, (short)0, acc1, false, false);
        acc2 = __builtin_amdgcn_wmma_f32_16x16x4_f32(false, a, false, b2, (short)0, acc2, false, false);
        acc3 = __builtin_amdgcn_wmma_f32_16x16x4_f32(false, a, false, b3, (short)0, acc3, false, false);
    }

    // Store D: VGPR v -> row (rb*16 + khalf*8 + v), column m within each 16-col tile
    const int mbase = rb * 16 + khalf * 8;
    float* __restrict__ op = out + (size_t)mbase * OUT_C + m;
    #pragma unroll
    for (int v = 0; v < 8; ++v) {
        if (mbase + v < N) {
            op[(size_t)v * OUT_C +  0] = acc0[v];
            op[(size_t)v * OUT_C + 16] = acc1[v];
            op[(size_t)v * OUT_C + 32] = acc2[v];
            op[(size_t)v * OUT_C + 48] = acc3[v];
        }
    }
}

// ---------------------------------------------------------------------------
// Kernel 2: zero a buffer (float4 granularity)
// ---------------------------------------------------------------------------
__global__ __launch_bounds__(256) void ngcn_zero_f4(v4f* __restrict__ p, long n4)
{
    long i = (long)blockIdx.x * blockDim.x + threadIdx.x;
    if (i < n4) p[i] = (v4f){};
}

// ---------------------------------------------------------------------------
// Kernel 3: one sparse propagation  dst[row] += val * src[col]
// One wave per edge; each lane owns a contiguous float2 channel pair
// (32 lanes * 8B = 256B coalesced gather). Scatter via global_atomic_add_f32.
// ---------------------------------------------------------------------------
__global__ __launch_bounds__(256) void ngcn_prop(
    const int* __restrict__ rows, const int* __restrict__ cols,
    const float* __restrict__ vals, const float* __restrict__ src,
    float* __restrict__ dst, int E)
{
    long gid  = (long)blockIdx.x * blockDim.x + threadIdx.x;
    int  edge = (int)(gid >> 5);
    int  lane = (int)(gid & 31);
    if (edge >= E) return;

    const int   r = rows[edge];
    const int   c = cols[edge];
    const float v = vals[edge];

    v2f s = *(const v2f*)(src + (size_t)c * OUT_C + 2 * lane);
    float* dp = dst + (size_t)r * OUT_C + 2 * lane;
    atomicAdd(dp + 0, v * s.x);
    atomicAdd(dp + 1, v * s.y);
}

// ---------------------------------------------------------------------------
// Kernel 4: broadcast bias into out (out[n*64 + c] = bias[c])
// ---------------------------------------------------------------------------
__global__ __launch_bounds__(256) void ngcn_bias_init(
    float* __restrict__ out, const float* __restrict__ bias, long total)
{
    long i = (long)blockIdx.x * blockDim.x + threadIdx.x;
    if (i < total) out[i] = bias[i & (OUT_C - 1)];
}

// ---------------------------------------------------------------------------
// Launch: out = A*(A*(F@W)) + bias   (ITERS = 3 -> 2 propagations)
// ws needs N*OUT_C floats (25.6 MB) for the ping-pong buffer.
// ---------------------------------------------------------------------------
extern "C" void kernel_launch(void* const* d_in, const int* in_sizes, int n_in,
                              void* d_out, int out_size, void* d_ws, size_t ws_size,
                              hipStream_t stream)
{
    const int*   indices  = (const int*)  d_in[0];   // (2, E) int32
    const float* values   = (const float*)d_in[1];   // (E,)
    const float* features = (const float*)d_in[2];   // (N, 128)
    const float* weight   = (const float*)d_in[3];   // (128, 64)
    const float* bias     = (const float*)d_in[4];   // (1, 64)
    float* out = (float*)d_out;
    float* ws  = (float*)d_ws;

    const int E = in_sizes[1];
    const int N = in_sizes[2] / IN_C;
    const int* rows = indices;         // indices[0]
    const int* cols = indices + E;     // indices[1]

    const int  nRowBlocks = (N + 15) / 16;
    const long total      = (long)N * OUT_C;
    const long pthreads   = (long)E * 32;

    // 1) out = F @ W   (fp32 WMMA)
    ngcn_gemm_wmma<<<(nRowBlocks + 3) / 4, 128, 0, stream>>>(
        features, weight, out, N, nRowBlocks);

    // 2) ws = 0
    ngcn_zero_f4<<<(unsigned)((total / 4 + 255) / 256), 256, 0, stream>>>(
        (v4f*)ws, total / 4);

    // 3) ws += scatter(out)
    ngcn_prop<<<(unsigned)((pthreads + 255) / 256), 256, 0, stream>>>(
        rows, cols, values, out, ws, E);

    // 4) out = bias
    ngcn_bias_init<<<(unsigned)((total + 255) / 256), 256, 0, stream>>>(
        out, bias, total);

    // 5) out += scatter(ws)   ->  A^2 (F@W) + bias
    ngcn_prop<<<(unsigned)((pthreads + 255) / 256), 256, 0, stream>>>(
        rows, cols, values, ws, out, E);
}